// LocallyConnected2d_41455024341460
// MI455X (gfx1250) — compile-verified
//
#include <hip/hip_runtime.h>

// LocallyConnected2d forward on gfx1250 (MI455X), fp32 WMMA path.
// Per pixel: GEMM M=O=32, N=B=16, K=C*9=288 via V_WMMA_F32_16X16X4_F32.
// HBM-bound: 167 MB @ 23.3 TB/s ~= 7.2 us floor. Weight stream (151 MB) is
// double-buffered into LDS with GLOBAL_LOAD_ASYNC_TO_LDS_B128 (ASYNCcnt);
// output leaves via GLOBAL_STORE_ASYNC_FROM_LDS_B128.

#define B_   16
#define C_   32
#define O_   32
#define IHW  64
#define KK2  9                     // KH*KW
#define CC_CHUNK 4                 // channels per LDS chunk
#define KCH  (CC_CHUNK * KK2)      // 36 k-values per chunk
#define PIX  8                     // pixels per workgroup (consecutive w)
#define RUNS (O_ * CC_CHUNK)       // 128 contiguous 288B weight runs / chunk
#define RUNF (PIX * KK2)           // 72 floats per run

typedef float v2f __attribute__((ext_vector_type(2)));
typedef float v8f __attribute__((ext_vector_type(8)));
typedef int   v4i __attribute__((__vector_size__(16)));   // matches builtin

#define AS1 __attribute__((address_space(1)))
#define AS3 __attribute__((address_space(3)))

#ifndef __has_builtin
#define __has_builtin(x) 0
#endif

#if __has_builtin(__builtin_amdgcn_global_load_async_to_lds_b128)
#define HAVE_ASYNC_LD 1
#else
#define HAVE_ASYNC_LD 0
#endif

#if __has_builtin(__builtin_amdgcn_global_store_async_from_lds_b128)
#define HAVE_ASYNC_ST 1
#else
#define HAVE_ASYNC_ST 0
#endif

#if __has_builtin(__builtin_amdgcn_s_wait_asynccnt)
#define WAIT_ASYNC(n) __builtin_amdgcn_s_wait_asynccnt(n)
#else
#define WAIT_ASYNC(n) asm volatile("s_wait_asynccnt %0" ::"i"(n) : "memory")
#endif

// Issue one weight chunk (32o x 4c x 8pix x 9j = 128 runs x 72 floats) into
// an LDS buffer mirroring the global run layout. 2304 b128 ops / 256 threads
// = 9 per thread, all lanes active (EXEC all ones). Runs are 16B aligned
// because pixbase is a multiple of 8 (=> run start byte offset % 16 == 0).
__device__ __forceinline__ void issue_w_chunk(const float* __restrict__ wgt,
                                              int cbase, int pixbase,
                                              float* __restrict__ wbuf,
                                              int tid) {
    for (int q = tid; q < RUNS * (RUNF / 4); q += 256) {
        const int run = q / 18;          // o*4 + cl
        const int t16 = q - run * 18;    // 16B chunk within run
        const int o   = run >> 2;
        const int cl  = run & 3;
        const float* gsrc = wgt +
            ((size_t)(o * C_ + cbase + cl) * (IHW * IHW) + pixbase) * KK2 +
            t16 * 4;
        float* ldst = wbuf + run * RUNF + t16 * 4;
#if HAVE_ASYNC_LD
        __builtin_amdgcn_global_load_async_to_lds_b128(
            (AS1 v4i*)gsrc, (AS3 v4i*)ldst, 0, 0);
#else
        *(float4*)ldst = *(const float4*)gsrc;
#endif
    }
}

__global__ __launch_bounds__(256) void lc2d_wmma_f32_kernel(
    const float* __restrict__ x,      // [B, C, 64, 64]
    const float* __restrict__ wgt,    // [O, C, 64, 64, 9]
    float* __restrict__ out)          // [B, O, 64, 64]
{
    const int wg   = blockIdx.x;            // 0..511
    const int h    = wg >> 3;               // 0..63
    const int w0   = (wg & 7) * PIX;        // 0,8,...,56
    const int pixbase = h * IHW + w0;       // multiple of 8 -> 16B-aligned runs

    const int tid    = threadIdx.x;
    const int waveid = tid >> 5;            // 0..7 -> pixel index
    const int lane   = tid & 31;
    const int half   = lane >> 4;           // 0: K{0,1}, 1: K{2,3}
    const int ml     = lane & 15;           // M index (A) / N index (B)

    // Weight double buffer mirrors global run layout (async b128 friendly).
    __shared__ __align__(16) float Wbuf[2][RUNS * RUNF];   // 2 x 36 KB
    // Patches pair-interleaved: [kk/2][pix][b][2] -> one 8B load per B-frag.
    __shared__ v2f Ppair[(KCH / 2) * PIX * B_];            // 18 KB
    __shared__ __align__(16) float Olds[O_ * B_ * PIX];    // 16 KB

    v8f acc0 = {0.f,0.f,0.f,0.f,0.f,0.f,0.f,0.f};
    v8f acc1 = {0.f,0.f,0.f,0.f,0.f,0.f,0.f,0.f};

    issue_w_chunk(wgt, 0, pixbase, Wbuf[0], tid);          // prologue: chunk 0

    for (int cc = 0; cc < C_ / CC_CHUNK; ++cc) {
        const int cbase = cc * CC_CHUNK;

        // ---- Stage patches P[kk][pix][b] (pair-interleaved), 18 iters ----
        float* Pf = (float*)Ppair;
        for (int idx = tid; idx < KCH * PIX * B_; idx += 256) {
            const int b  = idx & 15;
            const int p  = (idx >> 4) & 7;
            const int kk = idx >> 7;
            const int cl = kk / KK2;
            const int j  = kk - cl * KK2;
            const int di = j / 3, dj = j - di * 3;
            const int hh = h - 1 + di;
            const int ww = w0 + p - 1 + dj;
            float v = 0.f;
            if ((unsigned)hh < IHW && (unsigned)ww < IHW)
                v = x[((size_t)(b * C_ + cbase + cl) * IHW + hh) * IHW + ww];
            Pf[(((kk >> 1) * PIX + p) * B_ + b) * 2 + (kk & 1)] = v;
        }

        // ---- Prefetch next weight chunk async; wait for current chunk ----
        if (cc < 7) {
            issue_w_chunk(wgt, cbase + CC_CHUNK, pixbase, Wbuf[(cc + 1) & 1], tid);
            WAIT_ASYNC(9);   // async completes in-order => chunk cc is in LDS
        } else {
            WAIT_ASYNC(0);
        }
        __syncthreads();

        // ---- WMMA loop: this wave's pixel, 9 k-steps, 2 o-tiles ----
        const float* Wb = Wbuf[cc & 1];
        const int p = waveid;
        #pragma unroll
        for (int s = 0; s < KCH / 4; ++s) {
            const int kkA = s * 4 + 2 * half;   // even; lanes<16:K0/1, >=16:K2/3
            const v2f bf = Ppair[((2 * s + half) * PIX + p) * B_ + ml];

            const int cl0 = kkA / KK2, j0 = kkA - cl0 * KK2;
            const int kk1 = kkA + 1;
            const int cl1 = kk1 / KK2, j1 = kk1 - cl1 * KK2;
            const int pj  = p * KK2;

            v2f a0;
            a0.x = Wb[((ml) * 4 + cl0) * RUNF + pj + j0];
            a0.y = Wb[((ml) * 4 + cl1) * RUNF + pj + j1];
            acc0 = __builtin_amdgcn_wmma_f32_16x16x4_f32(
                       false, a0, false, bf, (short)0, acc0, false, false);

            v2f a1;
            a1.x = Wb[((ml + 16) * 4 + cl0) * RUNF + pj + j0];
            a1.y = Wb[((ml + 16) * 4 + cl1) * RUNF + pj + j1];
            acc1 = __builtin_amdgcn_wmma_f32_16x16x4_f32(
                       false, a1, false, bf, (short)0, acc1, false, false);
        }

        __syncthreads();   // all waves done reading Ppair/Wbuf[cc&1]
    }

    // ---- Transpose D through LDS so global stores are contiguous in w ----
    {
        const int p = waveid;
        #pragma unroll
        for (int r = 0; r < 8; ++r) {
            const int m  = r + 8 * half;
            Olds[((m)      * B_ + ml) * PIX + p] = acc0[r];
            Olds[((m + 16) * B_ + ml) * PIX + p] = acc1[r];
        }
    }
    __syncthreads();

    // 512 (o,b) pairs -> each thread writes 8 consecutive w as 2x 16B.
    for (int pi = tid; pi < O_ * B_; pi += 256) {
        const int o = pi >> 4;
        const int b = pi & 15;
        float* src = &Olds[(o * B_ + b) * PIX];
        float* dst = out + (size_t)(b * O_ + o) * (IHW * IHW) + pixbase;
#if HAVE_ASYNC_ST
        __builtin_amdgcn_global_store_async_from_lds_b128(
            (AS1 v4i*)dst,       (AS3 v4i*)src,       0, 0);
        __builtin_amdgcn_global_store_async_from_lds_b128(
            (AS1 v4i*)(dst + 4), (AS3 v4i*)(src + 4), 0, 0);
#else
        float4 lo = { src[0], src[1], src[2], src[3] };
        float4 hi = { src[4], src[5], src[6], src[7] };
        *(float4*)(dst)     = lo;
        *(float4*)(dst + 4) = hi;
#endif
    }
    // S_ENDPGM performs an implicit wait-idle, covering outstanding ASYNCcnt.
}

extern "C" void kernel_launch(void* const* d_in, const int* in_sizes, int n_in,
                              void* d_out, int out_size, void* d_ws, size_t ws_size,
                              hipStream_t stream) {
    (void)in_sizes; (void)n_in; (void)d_ws; (void)ws_size; (void)out_size;
    const float* x   = (const float*)d_in[0];   // (16,32,64,64) fp32
    const float* wgt = (const float*)d_in[1];   // (32,32,64,64,9) fp32
    float* out = (float*)d_out;                 // (16,32,64,64) fp32

    // 64 rows * 8 pixel-groups = 512 workgroups of 256 threads (8 wave32).
    lc2d_wmma_f32_kernel<<<dim3(512), dim3(256), 0, stream>>>(x, wgt, out);
}